// TrainiumMatMul_38276748542446
// MI455X (gfx1250) — compile-verified
//
#include <hip/hip_runtime.h>

// fp32 GEMM: C[M,N] = A[M,K] * B[K,N], K = 64, via V_WMMA_F32_16X16X4_F32.
// Output-bandwidth-bound (256 MB of C stores vs 4 MB of L2-resident A/B), so
// fp32 WMMA keeps exact reference precision at no performance cost.
//
// N is passed as a template constant for the reference shape (8192) so all
// per-k-group B addresses / C-store rows fold into 24-bit immediate offsets
// (strides 128KB / 32KB << 8.3MB limit), eliminating per-iteration 64-bit
// VALU address math.

typedef __attribute__((ext_vector_type(2))) float v2f;
typedef __attribute__((ext_vector_type(8))) float v8f;

#define K_DIM 64          // fixed by the reference
#define BLK_M 128         // block tile M (4 waves x 32)
#define BLK_N 128         // block tile N (2 waves x 64)

template <int NCONST>
__global__ __launch_bounds__(256, 2)
void gemm_f32_wmma_k64(const float* __restrict__ A,
                       const float* __restrict__ B,
                       float* __restrict__ C,
                       int Ndyn) {
  const int N = NCONST ? NCONST : Ndyn;

  const int lane = threadIdx.x & 31;
  const int wid  = threadIdx.x >> 5;   // 8 waves per block
  const int half = lane >> 4;          // 0: lanes 0-15, 1: lanes 16-31
  const int lo   = lane & 15;

  // 8 waves arranged 4 (M) x 2 (N); each wave computes a 32x64 region
  const int waveM = wid & 3;
  const int waveN = wid >> 2;
  const int wm = blockIdx.y * BLK_M + waveM * 32;  // wave M base (2 tiles of 16)
  const int wn = blockIdx.x * BLK_N + waveN * 64;  // wave N base (4 tiles of 16)

  // Hoisted base pointers; all per-iteration deltas become immediate offsets
  // when N is a compile-time constant.
  const int k0 = 2 * half;  // this lane-half's base K pair within a k-group
  const float* a0 = A + (size_t)(wm + lo) * K_DIM + k0;          // mt via +16*K_DIM
  const float* a1 = a0 + 16 * K_DIM;
  const float* b0 = B + (size_t)k0 * N + (wn + lo);              // nt via +16, kg via +4*N

  v8f acc[2][4] = {};  // 2 M-tiles x 4 N-tiles of 16x16 f32

  // K = 64 -> 16 k-groups of 4, fully unrolled.
  #pragma unroll
  for (int kg = 0; kg < K_DIM / 4; ++kg) {
    // A fragments: 16x4 tile; lane lo = row M, half selects K pair.
    // Row-major A (K-contiguous) -> one float2 (b64) load per lane,
    // advancing 16B per k-group (immediate offset).
    v2f a[2];
    a[0] = *(const v2f*)(a0 + 4 * kg);
    a[1] = *(const v2f*)(a1 + 4 * kg);

    // B fragments: 4x16 tile; lane lo = column N, half selects K pair.
    // Row-major B (stride N) -> two b32 loads per lane; kg stride = 4*N*4B
    // and k+1 stride = N*4B are immediate offsets for constant N.
    v2f b[4];
    #pragma unroll
    for (int nt = 0; nt < 4; ++nt) {
      const float* bp = b0 + (size_t)(4 * kg) * N + nt * 16;
      b[nt].x = bp[0];
      b[nt].y = bp[N];
    }

    // 8 WMMAs per k-group; A frag reused 4x, B frag reused 2x.
    #pragma unroll
    for (int mt = 0; mt < 2; ++mt) {
      #pragma unroll
      for (int nt = 0; nt < 4; ++nt) {
        acc[mt][nt] = __builtin_amdgcn_wmma_f32_16x16x4_f32(
            /*neg_a=*/false, a[mt],
            /*neg_b=*/false, b[nt],
            /*c_mod=*/(short)0, acc[mt][nt],
            /*reuse_a=*/false, /*reuse_b=*/false);
      }
    }
  }

  // Store: C/D layout = 8 VGPRs; row = wm + mt*16 + r + 8*half, col = wn + nt*16 + lo.
  // Each half-wave writes a contiguous 64B run; the wave's 4 n-tiles cover a
  // contiguous 256B per row so L2 merges full 128B lines. Row stride (N*4B)
  // is an immediate offset for constant N.
  float* c0 = C + (size_t)(wm + 8 * half) * N + (wn + lo);
  #pragma unroll
  for (int mt = 0; mt < 2; ++mt) {
    #pragma unroll
    for (int nt = 0; nt < 4; ++nt) {
      #pragma unroll
      for (int r = 0; r < 8; ++r) {
        c0[(size_t)(mt * 16 + r) * N + nt * 16] = acc[mt][nt][r];
      }
    }
  }
}

extern "C" void kernel_launch(void* const* d_in, const int* in_sizes, int n_in,
                              void* d_out, int out_size, void* d_ws, size_t ws_size,
                              hipStream_t stream) {
  const float* A = (const float*)d_in[0];  // (M, K) fp32
  const float* B = (const float*)d_in[1];  // (K, N) fp32
  float* C = (float*)d_out;                // (M, N) fp32

  const int M = in_sizes[0] / K_DIM;
  const int N = in_sizes[1] / K_DIM;

  dim3 grid((N + BLK_N - 1) / BLK_N, (M + BLK_M - 1) / BLK_M);
  dim3 block(256);
  if (N == 8192) {
    gemm_f32_wmma_k64<8192><<<grid, block, 0, stream>>>(A, B, C, N);
  } else {
    gemm_f32_wmma_k64<0><<<grid, block, 0, stream>>>(A, B, C, N);
  }
}